// weight_alter_56367150793463
// MI455X (gfx1250) — compile-verified
//
#include <hip/hip_runtime.h>
#include <math.h>

// ---------------------------------------------------------------------------
// weight_alter for MI455X (gfx1250, wave32).
// b=64, d=n=920, k2=32, sigma=0.1. One workgroup per batch b.
// Phase 1: per-sample perturbed top-32 (threefry RNG + wave argmax), sorted
//          indices -> packed u16 count histogram in LDS (exact indicators*n).
// Phase 2: s[k] = (counts[k,:] . wa[:]) / n via V_WMMA_F32_16X16X4_F32
//          (two 16-row M tiles, K loop of 230). B replicates wa across all
//          16 N columns so the loop body is branch-free (no exec toggling);
//          column 0 of D is read out. f32 WMMA keeps full precision.
// Phase 3: bbox transform -> out (64,32,4) f32.
// ---------------------------------------------------------------------------

typedef __attribute__((ext_vector_type(2))) float v2f;
typedef __attribute__((ext_vector_type(8))) float v8f;

#define B_DIM   64
#define D_DIM   920
#define N_DIM   920
#define K_SEL   32
#define SIGMA   0.1f
#define NPL     29          // ceil(920/32) values per lane
#define THREADS 512         // 16 waves (wave32)
#define HALFN   27084800u   // (64*920*920)/2, JAX threefry counter split

__device__ __forceinline__ unsigned rotl32(unsigned x, int r) {
  return (x << r) | (x >> (32 - r));
}

// threefry2x32 with key = jax.random.key(1) -> (0, 1)
__device__ __forceinline__ void threefry2x32(unsigned x0, unsigned x1,
                                             unsigned& o0, unsigned& o1) {
  const unsigned k0 = 0u, k1 = 1u;
  const unsigned k2 = k0 ^ k1 ^ 0x1BD11BDAu;
  x0 += k0; x1 += k1;
#define TF(r) { x0 += x1; x1 = rotl32(x1, (r)); x1 ^= x0; }
  TF(13) TF(15) TF(26) TF(6)   x0 += k1; x1 += k2 + 1u;
  TF(17) TF(29) TF(16) TF(24)  x0 += k2; x1 += k0 + 2u;
  TF(13) TF(15) TF(26) TF(6)   x0 += k0; x1 += k1 + 3u;
  TF(17) TF(29) TF(16) TF(24)  x0 += k1; x1 += k2 + 4u;
  TF(13) TF(15) TF(26) TF(6)   x0 += k2; x1 += k0 + 5u;
#undef TF
  o0 = x0; o1 = x1;
}

// jax.random.normal element e of the flat (b,n,d) tensor
__device__ __forceinline__ float jax_normal(unsigned e) {
  bool second = (e >= HALFN);
  unsigned c0 = second ? (e - HALFN) : e;
  unsigned o0, o1;
  threefry2x32(c0, c0 + HALFN, o0, o1);
  unsigned bits = second ? o1 : o0;
  float f = __uint_as_float(0x3f800000u | (bits >> 9)) - 1.0f;  // [0,1)
  const float lo = -0.99999994f;                                // nextafter(-1,0)
  float u = fmaxf(lo, f * (1.0f - lo) + lo);                    // [-1,1)
  return 1.41421356237f * erfinvf(u);
}

__global__ __launch_bounds__(THREADS)
void weight_alter_56367150793463_kernel(const float* __restrict__ weight_attn,
                                        float* __restrict__ out) {
  __shared__ unsigned cnt[K_SEL][D_DIM / 2];   // packed u16 counts [k][d/2]
  __shared__ float    wa_s[D_DIM];
  __shared__ float    s_s[K_SEL];

  const int b    = blockIdx.x;
  const int tid  = threadIdx.x;
  const int lane = tid & 31;
  const int wave = tid >> 5;

  for (int i = tid; i < K_SEL * (D_DIM / 2); i += THREADS)
    (&cnt[0][0])[i] = 0u;
  for (int i = tid; i < D_DIM; i += THREADS)
    wa_s[i] = weight_attn[b * D_DIM + i];
  __syncthreads();

  // ---------------- Phase 1: one wave per sample n ----------------
  for (int n = wave; n < N_DIM; n += THREADS / 32) {
    float val[NPL];
    const unsigned ebase = ((unsigned)b * N_DIM + (unsigned)n) * D_DIM;
#pragma unroll
    for (int j = 0; j < NPL; ++j) {
      int d = lane + 32 * j;
      val[j] = (d < D_DIM) ? (wa_s[d] + SIGMA * jax_normal(ebase + (unsigned)d))
                           : -INFINITY;
    }
    float maxv = -INFINITY; int maxj = 0;
#pragma unroll
    for (int j = 0; j < NPL; ++j)
      if (val[j] > maxv) { maxv = val[j]; maxj = j; }
    int maxd = lane + 32 * maxj;

    int selIdx = 0;
    for (int r = 0; r < K_SEL; ++r) {
      float bv = maxv; int bd = maxd;
#pragma unroll
      for (int off = 16; off > 0; off >>= 1) {   // wave argmax (tie -> low d)
        float ov = __shfl_xor(bv, off, 32);
        int   od = __shfl_xor(bd, off, 32);
        if (ov > bv || (ov == bv && od < bd)) { bv = ov; bd = od; }
      }
      if (lane == r) selIdx = bd;                // rank r winner
      if (bd == maxd) {                          // this lane owned the winner
#pragma unroll
        for (int j = 0; j < NPL; ++j) if (j == maxj) val[j] = -INFINITY;
        maxv = -INFINITY; maxj = 0;
#pragma unroll
        for (int j = 0; j < NPL; ++j)
          if (val[j] > maxv) { maxv = val[j]; maxj = j; }
        maxd = lane + 32 * maxj;
      }
    }

    // bitonic ascending sort of the 32 winning indices across the wave
#pragma unroll
    for (int k = 2; k <= 32; k <<= 1) {
#pragma unroll
      for (int j = k >> 1; j > 0; j >>= 1) {
        int  other = __shfl_xor(selIdx, j, 32);
        bool up    = ((lane & k) == 0);
        bool upper = ((lane & j) != 0);
        selIdx = (upper == up) ? max(selIdx, other) : min(selIdx, other);
      }
    }

    // counts <= 920 < 65535: halfword add never carries across the packing
    atomicAdd(&cnt[lane][selIdx >> 1], (selIdx & 1) ? 65536u : 1u);
  }
  __syncthreads();

  // ------- Phase 2: s = counts(32x920) . wa(920) via f32 WMMA -------
  if (wave < 2) {
    const int m  = 16 * wave + (lane & 15);  // A row (M)
    const int kp = lane >> 4;                // 0: K={4t,4t+1}, 1: K={4t+2,4t+3}
    v8f acc = {};
    for (int t = 0; t < D_DIM / 4; ++t) {
      unsigned cw = cnt[m][2 * t + kp];
      v2f A; A.x = (float)(cw & 0xFFFFu); A.y = (float)(cw >> 16);
      // Replicate wa across every N column (branch-free broadcast load);
      // all columns of D then hold s, we read column 0.
      v2f Bm;
      Bm.x = wa_s[4 * t + 2 * kp];
      Bm.y = wa_s[4 * t + 2 * kp + 1];
      acc = __builtin_amdgcn_wmma_f32_16x16x4_f32(
          /*neg_a=*/false, A, /*neg_b=*/false, Bm,
          /*c_mod=*/(short)0, acc, /*reuse_a=*/false, /*reuse_b=*/false);
    }
    // D layout: lanes 0-15 -> N=lane, M=r; lanes 16-31 -> N=lane-16, M=8+r
    if (lane == 0) {
#pragma unroll
      for (int r = 0; r < 8; ++r) s_s[16 * wave + r] = acc[r];
    } else if (lane == 16) {
#pragma unroll
      for (int r = 0; r < 8; ++r) s_s[16 * wave + 8 + r] = acc[r];
    }
  }
  __syncthreads();

  // ---------------- Phase 3: bbox ----------------
  if (tid < K_SEL) {
    float s  = s_s[tid] * (1.0f / (float)N_DIM);
    float r  = floorf(s / 40.0f);
    float c  = s - 40.0f * r;                       // jnp.mod(s, 40)
    float x0 = (c < 1.0f ? c : c - 1.0f) * 32.0f;
    float y0 = (r < 1.0f ? r : r - 1.0f) * 32.0f;
    float x1 = (c < 1.0f ? c + 2.0f : c + 1.0f) * 32.0f;
    float y1 = (r + 2.0f) * 32.0f;
    float* o = out + ((size_t)b * K_SEL + tid) * 4;
    o[0] = x0; o[1] = y0; o[2] = x1; o[3] = y1;
  }
}

extern "C" void kernel_launch(void* const* d_in, const int* in_sizes, int n_in,
                              void* d_out, int out_size, void* d_ws, size_t ws_size,
                              hipStream_t stream) {
  (void)in_sizes; (void)n_in; (void)d_ws; (void)ws_size; (void)out_size;
  // d_in[0]=input (unused), d_in[1]=src (unused), d_in[2]=weight_attn,
  // d_in[3]=k2 (==32, baked into K_SEL)
  const float* weight_attn = (const float*)d_in[2];
  float* out = (float*)d_out;
  weight_alter_56367150793463_kernel<<<dim3(B_DIM), dim3(THREADS), 0, stream>>>(
      weight_attn, out);
}